// CTCLayer_51015621542045
// MI455X (gfx1250) — compile-verified
//
#include <hip/hip_runtime.h>
#include <hip/hip_bf16.h>

// ---------------------------------------------------------------------------
// CTC loss (keras ctc_batch_cost style) for B=32, T=2048, C=96, L=256.
// Kernel A: logp = log(softmax(logits)+eps) -> d_ws (24 MB, L2-resident).
// Kernel B: serial alpha DP, 1 block per batch, alpha in LDS double buffer,
//           logp rows streamed into an LDS ring via CDNA5 async-to-LDS DMA.
// ---------------------------------------------------------------------------

#define WIDTH_DOWN 8
#define NEGF (-1e30f)
#define EPSF (1e-7f)

constexpr int B_ = 32;
constexpr int T_ = 2048;
constexpr int C_ = 96;
constexpr int L_ = 256;
constexpr int S_ = 2 * L_ + 1;      // 513 extended states
constexpr int BLANK_ = C_ - 1;      // 95

constexpr int NS_ = 8;              // LDS ring slots for logp rows
constexpr int ID_ = 4;              // async prefetch depth (ID_ + 2 <= NS_)

// ---- CDNA5 async global->LDS DMA (ASYNCcnt) -------------------------------
#if defined(__has_builtin)
#if __has_builtin(__builtin_amdgcn_global_load_async_to_lds_b128)
#define USE_ASYNC_LDS 1
#if __has_builtin(__builtin_amdgcn_s_wait_asynccnt)
#define WAIT_ASYNC(n) __builtin_amdgcn_s_wait_asynccnt(n)
#else
#define WAIT_ASYNC(n) asm volatile("s_wait_asynccnt %0" ::"i"(n) : "memory")
#endif
#endif
#endif

#ifdef USE_ASYNC_LDS
typedef int v4i_ __attribute__((vector_size(16)));
typedef __attribute__((address_space(1))) v4i_* gv4_t;  // global v4i*
typedef __attribute__((address_space(3))) v4i_* lv4_t;  // LDS v4i*
__device__ __forceinline__ void async_copy16(void* lds_dst, const float* gsrc) {
  __builtin_amdgcn_global_load_async_to_lds_b128(
      (gv4_t)(void*)const_cast<float*>(gsrc), (lv4_t)lds_dst, 0, 0);
}
#endif

// logaddexp on hardware v_exp_f32 / v_log_f32
__device__ __forceinline__ float lae(float a, float b) {
  float mx = fmaxf(a, b);
  float d  = fminf(a, b) - mx;  // <= 0 (0 when both equal, incl. both NEG)
  return mx + __logf(1.0f + __expf(d));
}

// ---------------------------------------------------------------------------
// Kernel A: per-(b,t) row of 96: logp = log(softmax + eps).
// One wave32 per row; 3 elements per lane; shfl_xor reductions.
// ---------------------------------------------------------------------------
__global__ __launch_bounds__(256) void ctc_logp_kernel(
    const float* __restrict__ logits, float* __restrict__ logp) {
  const int lane = threadIdx.x & 31;
  const int wid  = threadIdx.x >> 5;
  const int row  = blockIdx.x * 8 + wid;  // [0, B_*T_)
  const float* src = logits + (size_t)row * C_;
  float x0 = src[lane], x1 = src[lane + 32], x2 = src[lane + 64];

  float m = fmaxf(fmaxf(x0, x1), x2);
#pragma unroll
  for (int off = 16; off; off >>= 1) m = fmaxf(m, __shfl_xor(m, off));

  float e0 = __expf(x0 - m), e1 = __expf(x1 - m), e2 = __expf(x2 - m);
  float z = e0 + e1 + e2;
#pragma unroll
  for (int off = 16; off; off >>= 1) z += __shfl_xor(z, off);

  // log(exp(x-m)/z + eps) = log(exp(x-m) + eps*z) - log(z)
  const float lz = __logf(z);
  const float ez = EPSF * z;
  float* dst = logp + (size_t)row * C_;
  dst[lane]      = __logf(e0 + ez) - lz;
  dst[lane + 32] = __logf(e1 + ez) - lz;
  dst[lane + 64] = __logf(e2 + ez) - lz;
}

// ---------------------------------------------------------------------------
// Kernel B: alpha DP. One block per batch; thread s owns state s.
// ---------------------------------------------------------------------------
__global__ __launch_bounds__(544) void ctc_dp_kernel(
    const int* __restrict__ labels, const int* __restrict__ widths,
    const int* __restrict__ lengths, const float* __restrict__ logp,
    float* __restrict__ out) {
  const int b = blockIdx.x;
  const int s = threadIdx.x;

  __shared__ float alpha[2][S_ + 3];  // [buf][2 pad + 513 states]
#ifdef USE_ASYNC_LDS
  __shared__ float lp[NS_][C_];       // ring of logp rows
#endif

  const float* rowbase = logp + (size_t)b * T_ * C_;
  const int input_len = widths[b] / WIDTH_DOWN;  // in [3T/4, T]
  const int Tlast = input_len - 1;               // last updated step

  // Extended symbol + skip-transition flag for state s (computed once).
  int  ext  = BLANK_;
  bool skip = false;
  if (s < S_) {
    if (s & 1) {
      const int li = s >> 1;
      ext  = labels[b * L_ + li];
      skip = (ext != BLANK_) && ((s == 1) || (ext != labels[b * L_ + li - 1]));
    }
  }

  // alpha0: NEG except s=0,1 which take logp[b,0,ext[s]]; pad slots = NEG.
  if (s < 2) { alpha[0][s] = NEGF; alpha[1][s] = NEGF; }
  if (s < S_) alpha[0][s + 2] = (s <= 1) ? rowbase[ext] : NEGF;

#ifdef USE_ASYNC_LDS
  // Prologue: wave 0 lanes 0..23 DMA rows 1..ID_ (16B each lane, one
  // instruction per row -> ASYNCcnt increments once per row for wave 0).
  if (s < 24) {
#pragma unroll
    for (int r = 1; r <= ID_; ++r) {
      if (r <= Tlast)
        async_copy16(&lp[r & (NS_ - 1)][s * 4],
                     rowbase + (size_t)r * C_ + s * 4);
    }
  }
#endif

  for (int t = 1; t <= Tlast; ++t) {
    const int cur = (t - 1) & 1;
    const int nxt = t & 1;

#ifdef USE_ASYNC_LDS
    if (threadIdx.x < 32) {  // wave 0 only: issue row t+ID_, then wait row t
      const int r = t + ID_;
      if (r <= Tlast) {
        if (threadIdx.x < 24)
          async_copy16(&lp[r & (NS_ - 1)][threadIdx.x * 4],
                       rowbase + (size_t)r * C_ + threadIdx.x * 4);
        WAIT_ASYNC(ID_);     // row t complete; rows t+1..t+ID_ in flight
      } else {
        WAIT_ASYNC(0);       // drain tail
      }
    }
#else
    if (s < 3)  // pull row t+ID_ toward the WGP cache (global_prefetch_b8)
      __builtin_prefetch(rowbase + (size_t)(t + ID_) * C_ + s * 32, 0, 0);
#endif

    // Publishes: step t-1 alpha writes AND (via wave 0's wait) lp row t.
    __syncthreads();

    if (s < S_) {
      const float a0 = alpha[cur][s + 2];
      const float a1 = alpha[cur][s + 1];
      const float a2 = skip ? alpha[cur][s] : NEGF;
#ifdef USE_ASYNC_LDS
      const float lpv = lp[t & (NS_ - 1)][ext];
#else
      const float lpv = rowbase[(size_t)t * C_ + ext];  // L2-resident gather
#endif
      alpha[nxt][s + 2] = lae(lae(a0, a1), a2) + lpv;
    }
  }

  __syncthreads();
  if (s == 0) {
    const int fin = Tlast & 1;          // buffer written by the last step
    const int se  = 2 * lengths[b];     // last blank position
    const float ae  = alpha[fin][se + 2];
    const float ae1 = alpha[fin][se + 1];
    out[b] = -lae(ae, ae1);
  }
}

// ---------------------------------------------------------------------------
extern "C" void kernel_launch(void* const* d_in, const int* in_sizes, int n_in,
                              void* d_out, int out_size, void* d_ws,
                              size_t ws_size, hipStream_t stream) {
  (void)in_sizes; (void)n_in; (void)out_size; (void)ws_size;
  const int*   labels  = (const int*)d_in[0];
  const float* logits  = (const float*)d_in[1];
  const int*   widths  = (const int*)d_in[2];
  const int*   lengths = (const int*)d_in[3];
  float* outp = (float*)d_out;
  float* logp = (float*)d_ws;  // B*T*C floats = 24 MB scratch

  ctc_logp_kernel<<<(B_ * T_) / 8, 256, 0, stream>>>(logits, logp);
  ctc_dp_kernel<<<B_, 544, 0, stream>>>(labels, widths, lengths, logp, outp);
}